// HOPECore_49615462203740
// MI455X (gfx1250) — compile-verified
//
#include <hip/hip_runtime.h>
#include <hip/hip_bf16.h>
#include <math.h>

typedef float v2f __attribute__((ext_vector_type(2)));
typedef float v8f __attribute__((ext_vector_type(8)));
typedef int   v4i_vs __attribute__((vector_size(16)));   // b128 payload type

#define EPI_NONE        0
#define EPI_RELU        1
#define EPI_SIGMOID     2
#define EPI_TANH_MUL    3
#define EPI_TANH_ADDRES 4
#define EPI_ADDIN       5
#define EPI_GATE        6
#define EPI_DIAG        7

#if defined(__has_builtin)
#if __has_builtin(__builtin_amdgcn_global_load_async_to_lds_b128) && \
    __has_builtin(__builtin_amdgcn_s_wait_asynccnt)
#define USE_ASYNC_LDS 1
#endif
#endif
#ifndef USE_ASYNC_LDS
#define USE_ASYNC_LDS 0
#endif

#if USE_ASYNC_LDS
__device__ __forceinline__ void async_ld128(const float* g, float* l)
{
    // Builtin signature (from hipcc diagnostic): param0 is a pointer to a
    // 16-byte int vector in the __device__ (addrspace(1)) AS; param1 is the
    // LDS (addrspace(3)) destination; then imm offset, imm cpol.
    __builtin_amdgcn_global_load_async_to_lds_b128(
        (__attribute__((address_space(1))) v4i_vs*)g,
        (__attribute__((address_space(3))) v4i_vs*)l,
        0, 0);
}
#endif

// ---------------------------------------------------------------------------
// f32 WMMA GEMM:  C[M,N] = epilogue( A[M,K] @ W[N,K]^T + bias )
// A given as up to 3 row-major K-segments (fused concat inputs):
//   k in [0,k1) -> A0 (ld lda0), [k1,k2) -> A1, [k2,K) -> A2
// Block tile 128x128, BK=16, 256 threads (8 wave32), wave tile 32x64
// (2x4 grid of 16x16 WMMA accumulators).
// Double-buffered LDS pipeline: tile k+1 is moved global->LDS (async-to-LDS
// on gfx1250, no VGPR transit => no spills) while tile k is computed.
// One barrier per K-tile.  M,N multiples of 128; K,k1,k2 multiples of 16.
// ---------------------------------------------------------------------------
template<int MODE>
__global__ __launch_bounds__(256)
void gemm_wmma_f32(
    int M, int N, int K,
    const float* __restrict__ A0, const float* __restrict__ A1,
    const float* __restrict__ A2,
    int k1, int k2, int lda0, int lda1, int lda2,
    const float* __restrict__ W, int ldw,
    const float* __restrict__ bias,
    const float* __restrict__ P0, const float* __restrict__ P1,
    const float* __restrict__ P2,
    float* __restrict__ C, int ldc)
{
    __shared__ float As[2][128][20];   // stride 20: conflict-free, 16B-aligned
    __shared__ float Bs[2][128][20];   // 40 KB total (double-buffered)

    const int t    = threadIdx.x;
    const int lane = t & 31;
    const int wv   = t >> 5;
    const int wm   = (wv >> 1) << 5;   // wave row offset: 0,32,64,96
    const int wn   = (wv & 1) << 6;    // wave col offset: 0,64
    const int lm   = lane & 15;
    const int koff = (lane >> 4) << 1; // lanes 0-15 -> K{0,1}; 16-31 -> K{2,3}

    const int bx = blockIdx.x;         // N tile
    const int by = blockIdx.y;         // M tile
    const int rowL = t >> 2;           // 0..63  (loader row; +64 for chunk 1)
    const int c4   = (t & 3) << 2;     // 0,4,8,12 (loader K offset, float4)

    // Per-segment base pointers for this thread's loader lane
    const float* pa0 = A0 + (size_t)(by * 128 + rowL) * lda0 + c4;
    const float* pa1 = A1 + (size_t)(by * 128 + rowL) * lda1 + c4;
    const float* pa2 = A2 + (size_t)(by * 128 + rowL) * lda2 + c4;
    const float* pb  = W  + (size_t)(bx * 128 + rowL) * ldw  + c4;

    auto seg = [&](int kk0, const float** pp, int* ld) {
        if (kk0 < k1)      { *pp = pa0 + kk0;        *ld = lda0; }
        else if (kk0 < k2) { *pp = pa1 + (kk0 - k1); *ld = lda1; }
        else               { *pp = pa2 + (kk0 - k2); *ld = lda2; }
    };

    v8f acc[2][4];
#pragma unroll
    for (int i = 0; i < 2; ++i)
#pragma unroll
        for (int j = 0; j < 4; ++j)
#pragma unroll
            for (int r = 0; r < 8; ++r) acc[i][j][r] = 0.f;

    // ---- compute one BK=16 tile from LDS buffer `cur` -----------------
    auto compute_tile = [&](int cur) {
#pragma unroll
        for (int kk = 0; kk < 16; kk += 4) {
            v2f a[2], b[4];
#pragma unroll
            for (int i = 0; i < 2; ++i) {
                a[i].x = As[cur][wm + 16 * i + lm][kk + koff];
                a[i].y = As[cur][wm + 16 * i + lm][kk + koff + 1];
            }
#pragma unroll
            for (int j = 0; j < 4; ++j) {
                b[j].x = Bs[cur][wn + 16 * j + lm][kk + koff];
                b[j].y = Bs[cur][wn + 16 * j + lm][kk + koff + 1];
            }
#pragma unroll
            for (int i = 0; i < 2; ++i)
#pragma unroll
                for (int j = 0; j < 4; ++j)
                    acc[i][j] = __builtin_amdgcn_wmma_f32_16x16x4_f32(
                        false, a[i], false, b[j], (short)0, acc[i][j],
                        false, false);
        }
    };

#if USE_ASYNC_LDS
    // ---- gfx1250 async global->LDS pipeline (no VGPR staging) ---------
    auto issue_tile = [&](int kk0, int buf) {
        const float* pa; int ldsel;
        seg(kk0, &pa, &ldsel);
#pragma unroll
        for (int i = 0; i < 2; ++i)
            async_ld128(pa + (size_t)(64 * ldsel) * i,
                        &As[buf][rowL + 64 * i][c4]);
        const float* pbk = pb + kk0;
#pragma unroll
        for (int i = 0; i < 2; ++i)
            async_ld128(pbk + (size_t)(64 * ldw) * i,
                        &Bs[buf][rowL + 64 * i][c4]);
    };

    issue_tile(0, 0);
    int cur = 0;
    for (int k0 = 0;;) {
        __builtin_amdgcn_s_wait_asynccnt(0);  // my loads into LDS buffer cur done
        __syncthreads();                      // everyone's loads done; prev compute done
        const int kn = k0 + 16;
        if (kn < K) issue_tile(kn, cur ^ 1);  // overlaps with compute below
        compute_tile(cur);
        if (kn >= K) break;
        k0 = kn; cur ^= 1;
    }
#else
    // ---- fallback: short-lived register staging, stores after compute -
    float4 ra[2], rb[2];
    auto load_regs = [&](int kk0) {
        const float* pa; int ldsel;
        seg(kk0, &pa, &ldsel);
#pragma unroll
        for (int i = 0; i < 2; ++i)
            ra[i] = *(const float4*)(pa + (size_t)(64 * ldsel) * i);
        const float* pbk = pb + kk0;
#pragma unroll
        for (int i = 0; i < 2; ++i)
            rb[i] = *(const float4*)(pbk + (size_t)(64 * ldw) * i);
    };
    auto store_regs = [&](int buf) {
#pragma unroll
        for (int i = 0; i < 2; ++i)
            *(float4*)&As[buf][rowL + 64 * i][c4] = ra[i];
#pragma unroll
        for (int i = 0; i < 2; ++i)
            *(float4*)&Bs[buf][rowL + 64 * i][c4] = rb[i];
    };

    load_regs(0);
    store_regs(0);
    int cur = 0;
    for (int k0 = 0;;) {
        __syncthreads();                      // LDS buffer cur visible; prev compute done
        const int kn = k0 + 16;
        const bool more = kn < K;
        if (more) load_regs(kn);              // in flight during compute
        compute_tile(cur);
        if (!more) break;
        store_regs(cur ^ 1);                  // after compute: latency fully hidden
        k0 = kn; cur ^= 1;
    }
#endif

    // Epilogue. C/D layout: VGPR r -> row +r (lanes 0-15) / +8+r (lanes 16-31),
    // col = lane%16 within each 16x16 tile.
    const int rb2 = (lane >> 4) << 3;
#pragma unroll
    for (int i = 0; i < 2; ++i) {
#pragma unroll
        for (int j = 0; j < 4; ++j) {
            const int cc = bx * 128 + wn + 16 * j + lm;
            const float badd = bias ? bias[cc] : 0.f;
#pragma unroll
            for (int r = 0; r < 8; ++r) {
                const int gm = by * 128 + wm + 16 * i + rb2 + r;
                float v = acc[i][j][r] + badd;
                const size_t idx = (size_t)gm * ldc + cc;
                if (MODE == EPI_NONE) {
                    C[idx] = v;
                } else if (MODE == EPI_RELU) {
                    C[idx] = fmaxf(v, 0.f);
                } else if (MODE == EPI_SIGMOID) {
                    C[idx] = 1.f / (1.f + __expf(-v));
                } else if (MODE == EPI_TANH_MUL) {
                    C[idx] = tanhf(v) * P0[idx];
                } else if (MODE == EPI_TANH_ADDRES) {
                    C[idx] = P0[idx] + tanhf(v);
                } else if (MODE == EPI_ADDIN) {
                    C[idx] = v + P0[idx];
                } else if (MODE == EPI_GATE) {
                    const float g = P1[idx];
                    C[idx] = P0[idx] + g * P2[idx] + (1.f - g) * v;
                } else if (MODE == EPI_DIAG) {
                    C[idx] = v + ((gm == cc) ? 0.9f * tanhf(P0[gm]) : 0.f);
                }
            }
        }
    }
}

// ---------------------------------------------------------------------------
// Row LayerNorm over N=1024 (one block per row, 256 threads, float4 lanes).
// Population variance, eps = 1e-5 (matches jnp.var / torch LN default).
// ---------------------------------------------------------------------------
template<bool RELU>
__global__ __launch_bounds__(256)
void ln1024(const float* __restrict__ X, const float* __restrict__ G,
            const float* __restrict__ Bt, float* __restrict__ Y)
{
    __shared__ float s1[256];
    __shared__ float s2[256];
    const int row = blockIdx.x;
    const int t = threadIdx.x;
    float4 x = ((const float4*)(X + (size_t)row * 1024))[t];
    s1[t] = x.x + x.y + x.z + x.w;
    s2[t] = x.x * x.x + x.y * x.y + x.z * x.z + x.w * x.w;
    __syncthreads();
    for (int o = 128; o > 0; o >>= 1) {
        if (t < o) { s1[t] += s1[t + o]; s2[t] += s2[t + o]; }
        __syncthreads();
    }
    const float mean = s1[0] * (1.0f / 1024.0f);
    const float var  = s2[0] * (1.0f / 1024.0f) - mean * mean;
    const float rstd = rsqrtf(var + 1e-5f);
    float4 g = ((const float4*)G)[t];
    float4 b = ((const float4*)Bt)[t];
    float4 y;
    y.x = (x.x - mean) * rstd * g.x + b.x;
    y.y = (x.y - mean) * rstd * g.y + b.y;
    y.z = (x.z - mean) * rstd * g.z + b.z;
    y.w = (x.w - mean) * rstd * g.w + b.w;
    if (RELU) {
        y.x = fmaxf(y.x, 0.f); y.y = fmaxf(y.y, 0.f);
        y.z = fmaxf(y.z, 0.f); y.w = fmaxf(y.w, 0.f);
    }
    ((float4*)(Y + (size_t)row * 1024))[t] = y;
}

// ---------------------------------------------------------------------------
extern "C" void kernel_launch(void* const* d_in, const int* in_sizes, int n_in,
                              void* d_out, int out_size, void* d_ws, size_t ws_size,
                              hipStream_t stream)
{
    (void)in_sizes; (void)n_in; (void)out_size; (void)ws_size;

    const float* s_prev = (const float*)d_in[0];
    const float* w_prev = (const float*)d_in[1];
    const float* p_prev = (const float*)d_in[2];
    const float* e_t    = (const float*)d_in[3];
    const float* c_t    = (const float*)d_in[4];
    const float* fw1    = (const float*)d_in[5];
    const float* fb1    = (const float*)d_in[6];
    const float* fln_g  = (const float*)d_in[7];
    const float* fln_b  = (const float*)d_in[8];
    const float* fw2    = (const float*)d_in[9];
    const float* fb2    = (const float*)d_in[10];
    const float* A_diag = (const float*)d_in[11];
    const float* A_U    = (const float*)d_in[12];
    const float* A_V    = (const float*)d_in[13];
    const float* amod_w = (const float*)d_in[14];
    const float* amod_b = (const float*)d_in[15];
    const float* bnet_w = (const float*)d_in[16];
    const float* bnet_b = (const float*)d_in[17];
    const float* pw1    = (const float*)d_in[18];
    const float* pb1    = (const float*)d_in[19];
    const float* pln_g  = (const float*)d_in[20];
    const float* pln_b  = (const float*)d_in[21];
    const float* pw2    = (const float*)d_in[22];
    const float* pb2    = (const float*)d_in[23];
    const float* pw3    = (const float*)d_in[24];
    const float* pb3    = (const float*)d_in[25];
    const float* gw     = (const float*)d_in[26];
    const float* gb     = (const float*)d_in[27];
    const float* uw     = (const float*)d_in[28];
    const float* up     = (const float*)d_in[29];
    const float* oln_g  = (const float*)d_in[30];
    const float* oln_b  = (const float*)d_in[31];

    const int B = 8192;
    float* ws    = (float*)d_ws;
    float* Abase = ws;                          // 1024*1024
    float* bufA  = Abase + 1024 * 1024;         // B*1024 : y1/h -> y2/h1 -> s_pre
    float* bufB  = bufA + (size_t)B * 1024;     // B*512  : z -> h2
    float* bufC  = bufB + (size_t)B * 512;      // B*1024 : mod*w -> g
    float* bufD  = bufC + (size_t)B * 1024;     // B*1024 : w_linear -> t1

    float* out_s = (float*)d_out;               // [B,1024]
    float* out_w = out_s + (size_t)B * 1024;    // [B,1024]
    float* out_p = out_s + (size_t)B * 2048;    // [B,512]

    const dim3 blk(256);
    const float* Z = nullptr;
    const int GY = B / 128;                     // 64

    // 1. A_base = A_U @ A_V^T + diag(0.9*tanh(A_diag))   [1024,1024], K=256
    gemm_wmma_f32<EPI_DIAG><<<dim3(8, 8), blk, 0, stream>>>(
        1024, 1024, 256, A_U, A_U, A_U, 256, 256, 256, 256, 256,
        A_V, 256, Z, A_diag, Z, Z, Abase, 1024);

    // 2. y1 = [s||e||c] @ fw1^T + fb1                    [B,1024], K=2560
    gemm_wmma_f32<EPI_NONE><<<dim3(8, GY), blk, 0, stream>>>(
        B, 1024, 2560, s_prev, e_t, c_t, 1024, 2048, 1024, 1024, 512,
        fw1, 2560, fb1, Z, Z, Z, bufA, 1024);

    // 3. h = relu(LN(y1))   (in place)
    ln1024<true><<<B, blk, 0, stream>>>(bufA, fln_g, fln_b, bufA);

    // 4. z = h @ fw2^T + fb2                             [B,512], K=1024
    gemm_wmma_f32<EPI_NONE><<<dim3(4, GY), blk, 0, stream>>>(
        B, 512, 1024, bufA, bufA, bufA, 1024, 1024, 1024, 1024, 1024,
        fw2, 1024, fb2, Z, Z, Z, bufB, 512);

    // 5. mw = tanh(c @ amod^T + amod_b) * w_prev         [B,1024], K=512
    gemm_wmma_f32<EPI_TANH_MUL><<<dim3(8, GY), blk, 0, stream>>>(
        B, 1024, 512, c_t, c_t, c_t, 512, 512, 512, 512, 512,
        amod_w, 512, amod_b, w_prev, Z, Z, bufC, 1024);

    // 6. w_linear = mw @ A_base^T                        [B,1024], K=1024
    gemm_wmma_f32<EPI_NONE><<<dim3(8, GY), blk, 0, stream>>>(
        B, 1024, 1024, bufC, bufC, bufC, 1024, 1024, 1024, 1024, 1024,
        Abase, 1024, Z, Z, Z, Z, bufD, 1024);

    // 7. w_t = [c||z] @ bnet^T + bnet_b + w_linear  -> out_w
    gemm_wmma_f32<EPI_ADDIN><<<dim3(8, GY), blk, 0, stream>>>(
        B, 1024, 1024, c_t, bufB, bufB, 512, 1024, 512, 512, 512,
        bnet_w, 1024, bnet_b, bufD, Z, Z, out_w, 1024);

    // 8. y2 = [p||z||c] @ pw1^T + pb1                    [B,1024], K=1536
    gemm_wmma_f32<EPI_NONE><<<dim3(8, GY), blk, 0, stream>>>(
        B, 1024, 1536, p_prev, bufB, c_t, 512, 1024, 512, 512, 512,
        pw1, 1536, pb1, Z, Z, Z, bufA, 1024);

    // 9. h1 = relu(LN(y2))  (in place)
    ln1024<true><<<B, blk, 0, stream>>>(bufA, pln_g, pln_b, bufA);

    // 10. h2 = relu(h1 @ pw2^T + pb2)                    [B,512], K=1024
    gemm_wmma_f32<EPI_RELU><<<dim3(4, GY), blk, 0, stream>>>(
        B, 512, 1024, bufA, bufA, bufA, 1024, 1024, 1024, 1024, 1024,
        pw2, 1024, pb2, Z, Z, Z, bufB, 512);

    // 11. p_t = p_prev + tanh(h2 @ pw3^T + pb3)  -> out_p
    gemm_wmma_f32<EPI_TANH_ADDRES><<<dim3(4, GY), blk, 0, stream>>>(
        B, 512, 512, bufB, bufB, bufB, 512, 512, 512, 512, 512,
        pw3, 512, pb3, p_prev, Z, Z, out_p, 512);

    // 12. g = sigmoid([s||e||c] @ gw^T + gb)             [B,1024], K=2560
    gemm_wmma_f32<EPI_SIGMOID><<<dim3(8, GY), blk, 0, stream>>>(
        B, 1024, 2560, s_prev, e_t, c_t, 1024, 2048, 1024, 1024, 512,
        gw, 2560, gb, Z, Z, Z, bufC, 1024);

    // 13. t1 = p_t @ up^T                                [B,1024], K=512
    gemm_wmma_f32<EPI_NONE><<<dim3(8, GY), blk, 0, stream>>>(
        B, 1024, 512, out_p, out_p, out_p, 512, 512, 512, 512, 512,
        up, 512, Z, Z, Z, Z, bufD, 1024);

    // 14. s_pre = s_prev + g*t1 + (1-g)*(w_t @ uw^T)     [B,1024], K=1024
    gemm_wmma_f32<EPI_GATE><<<dim3(8, GY), blk, 0, stream>>>(
        B, 1024, 1024, out_w, out_w, out_w, 1024, 1024, 1024, 1024, 1024,
        uw, 1024, Z, s_prev, bufC, bufD, bufA, 1024);

    // 15. s_t = LN(s_pre) -> out_s
    ln1024<false><<<B, blk, 0, stream>>>(bufA, oln_g, oln_b, out_s);
}